// GraphAttentionLayer_35923106464274
// MI455X (gfx1250) — compile-verified
//
#include <hip/hip_runtime.h>
#include <hip/hip_bf16.h>
#include <stdint.h>

#define IN_F   128
#define OUT_F  64
#define EMBED  32
#define NNODE  8192
#define ALPHA  0.2f
#define CSLICE 8          // i-slices for column-stat pass
#define CHUNK  128        // j-rows of h staged in LDS per block iteration

typedef __attribute__((ext_vector_type(2))) float v2f;
typedef __attribute__((ext_vector_type(8))) float v8f;

// ---------------------------------------------------------------------------
// Kernel A: h = input @ W   (8192x128 @ 128x64 -> 8192x64). Tiny; plain VALU.
// ---------------------------------------------------------------------------
__global__ void gat_h_kernel(const float* __restrict__ input,
                             const float* __restrict__ W,
                             float* __restrict__ h) {
    int idx = blockIdx.x * blockDim.x + threadIdx.x;
    int row = idx >> 6;
    int col = idx & 63;
    const float* in = input + row * IN_F;
    float acc = 0.f;
#pragma unroll 8
    for (int k = 0; k < IN_F; ++k)
        acc = __builtin_fmaf(in[k], W[k * OUT_F + col], acc);
    h[idx] = acc;
}

// ---------------------------------------------------------------------------
// Kernel B: per-node scalars.
//   u[i] = h[i,:] . a[0:64]   v[i] = h[i,:] . a[64:128]
//   p[i] = beta * (emb[i,:] . intent_w[0:32])
// (beta*jj_j + beta*b is a per-column constant under softmax(axis=0): drops.)
// ---------------------------------------------------------------------------
__global__ void gat_vec_kernel(const float* __restrict__ h,
                               const float* __restrict__ emb,
                               const float* __restrict__ a,
                               const float* __restrict__ iw,
                               const float* __restrict__ beta_p,
                               float* __restrict__ u,
                               float* __restrict__ v,
                               float* __restrict__ p) {
    int i = blockIdx.x * blockDim.x + threadIdx.x;
    float beta = beta_p[0];
    const float* hr = h + i * OUT_F;
    float s = 0.f, d = 0.f;
#pragma unroll 8
    for (int k = 0; k < OUT_F; ++k) {
        float hv = hr[k];
        s = __builtin_fmaf(hv, a[k], s);
        d = __builtin_fmaf(hv, a[OUT_F + k], d);
    }
    const float* er = emb + i * EMBED;
    float pi = 0.f;
#pragma unroll 8
    for (int k = 0; k < EMBED; ++k)
        pi = __builtin_fmaf(er[k], iw[k], pi);
    u[i] = s;
    v[i] = d;
    p[i] = beta * pi;
}

// ---------------------------------------------------------------------------
// Kernel C1: partial per-column softmax stats over one i-slice.
//   f(i,j) = LReLU(u_i + v_j) + p_i ; online (m, Z) per (slice, j).
// grid = 32 j-blocks x CSLICE slices -> 2048 waves.
// ---------------------------------------------------------------------------
__global__ void __launch_bounds__(256)
gat_colstats_part(const float* __restrict__ u,
                  const float* __restrict__ p,
                  const float* __restrict__ v,
                  float* __restrict__ pm,
                  float* __restrict__ pz) {
    __shared__ float su[512];
    __shared__ float sp[512];
    int j     = (blockIdx.x & 31) * 256 + threadIdx.x;
    int slice = blockIdx.x >> 5;
    int ibeg  = slice * (NNODE / CSLICE);
    float vj = v[j];
    float m = -INFINITY, Z = 0.f;
    for (int i0 = ibeg; i0 < ibeg + NNODE / CSLICE; i0 += 512) {
        __syncthreads();
        su[threadIdx.x]       = u[i0 + threadIdx.x];
        su[threadIdx.x + 256] = u[i0 + threadIdx.x + 256];
        sp[threadIdx.x]       = p[i0 + threadIdx.x];
        sp[threadIdx.x + 256] = p[i0 + threadIdx.x + 256];
        __syncthreads();
#pragma unroll 4
        for (int t = 0; t < 512; ++t) {
            float x = su[t] + vj;
            x = x > 0.f ? x : ALPHA * x;
            float f = x + sp[t];
            if (f > m) { Z *= __expf(m - f); m = f; }
            Z += __expf(f - m);
        }
    }
    pm[slice * NNODE + j] = m;
    pz[slice * NNODE + j] = Z;
}

// ---------------------------------------------------------------------------
// Kernel C2: merge slice stats ->  r[j] = -(M_j + ln Z_j), P[i,j] <= 1.
// ---------------------------------------------------------------------------
__global__ void gat_colstats_merge(const float* __restrict__ pm,
                                   const float* __restrict__ pz,
                                   float* __restrict__ r) {
    int j = blockIdx.x * blockDim.x + threadIdx.x;
    float M = -INFINITY;
#pragma unroll
    for (int s = 0; s < CSLICE; ++s) M = fmaxf(M, pm[s * NNODE + j]);
    float Z = 0.f;
#pragma unroll
    for (int s = 0; s < CSLICE; ++s)
        Z += pz[s * NNODE + j] * __expf(pm[s * NNODE + j] - M);
    r[j] = -(M + __logf(Z));
}

// ---------------------------------------------------------------------------
// Kernel D: partial h' = P_slice @ h via V_WMMA_F32_16X16X4_F32.
// Block = 4 waves = 4 row-tiles sharing one j-slice. h/v/r chunks are staged
// into LDS with gfx1250 async global->LDS loads (ASYNCcnt), then B-operands
// come from ds_load. Each wave: A = 16x4 tile of P built in-register (2 exp
// per lane per step), 4 f32 accumulators cover the 64 output columns.
//   A: lane L holds A[L&15, klo..klo+1], klo=(L>=16)?2:0
//   B: VGPR0 = h[j0+klo, n], VGPR1 = h[j0+klo+1, n], n=L&15
//   C: VGPR q = row q (lanes 0-15) / q+8 (lanes 16-31)
// ---------------------------------------------------------------------------
__global__ void __launch_bounds__(128)
gat_attn_wmma_kernel(const float* __restrict__ h,
                     const float* __restrict__ u,
                     const float* __restrict__ p,
                     const float* __restrict__ v,
                     const float* __restrict__ r,
                     float* __restrict__ partial,
                     int jlen, int nd) {
    __shared__ float sh[CHUNK * OUT_F];   // 32 KB
    __shared__ float sv[CHUNK];
    __shared__ float sr[CHUNK];

    int tid   = threadIdx.x;
    int lane  = tid & 31;
    int wave  = tid >> 5;
    int rb    = blockIdx.x / nd;
    int slice = blockIdx.x % nd;
    int i0    = (rb * 4 + wave) * 16;
    int jbeg  = slice * jlen;

    int mrow = lane & 15;
    int klo  = (lane >> 4) << 1;          // 0 | 2
    int n    = lane & 15;

    float um  = u[i0 + mrow];
    float pmi = p[i0 + mrow];

    v8f c0 = {}, c1 = {}, c2 = {}, c3 = {};

    // LDS byte offsets: low 32 bits of the generic pointer (LDS aperture).
    unsigned lh = (unsigned)(uintptr_t)(void*)sh;
    unsigned lv = (unsigned)(uintptr_t)(void*)sv;
    unsigned lr = (unsigned)(uintptr_t)(void*)sr;

    for (int jc = jbeg; jc < jbeg + jlen; jc += CHUNK) {
        __syncthreads();   // previous chunk fully consumed
        // --- async stage: h[jc..jc+128) x 64  (16 x b128 per thread) ---
        const float* gsrc = h + (size_t)jc * OUT_F;
#pragma unroll
        for (int c = 0; c < (CHUNK * OUT_F) / (4 * 128); ++c) {
            int e4 = tid + c * 128;
            const float* g = gsrc + e4 * 4;
            unsigned l = lh + e4 * 16;
            asm volatile("global_load_async_to_lds_b128 %0, %1, off"
                         :: "v"(l), "v"(g) : "memory");
        }
        // --- async stage: v,r chunks (1 x b32 per thread each) ---
        {
            const float* gv = v + jc + tid;
            const float* gr = r + jc + tid;
            unsigned lv1 = lv + tid * 4;
            unsigned lr1 = lr + tid * 4;
            asm volatile("global_load_async_to_lds_b32 %0, %1, off"
                         :: "v"(lv1), "v"(gv) : "memory");
            asm volatile("global_load_async_to_lds_b32 %0, %1, off"
                         :: "v"(lr1), "v"(gr) : "memory");
        }
        asm volatile("s_wait_asynccnt 0x0" ::: "memory");
        __syncthreads();

#pragma unroll 4
        for (int jj = 0; jj < CHUNK; jj += 4) {
            int ja = jj + klo;
            int jb = ja + 1;

            float xa = um + sv[ja];
            xa = xa > 0.f ? xa : ALPHA * xa;
            float xb = um + sv[jb];
            xb = xb > 0.f ? xb : ALPHA * xb;

            v2f A;
            A.x = __expf(xa + pmi + sr[ja]);
            A.y = __expf(xb + pmi + sr[jb]);

            const float* ha = sh + ja * OUT_F + n;
            const float* hb = sh + jb * OUT_F + n;
            v2f B0 = { ha[0],  hb[0]  };
            v2f B1 = { ha[16], hb[16] };
            v2f B2 = { ha[32], hb[32] };
            v2f B3 = { ha[48], hb[48] };

            c0 = __builtin_amdgcn_wmma_f32_16x16x4_f32(false, A, false, B0, (short)0, c0, false, false);
            c1 = __builtin_amdgcn_wmma_f32_16x16x4_f32(false, A, false, B1, (short)0, c1, false, false);
            c2 = __builtin_amdgcn_wmma_f32_16x16x4_f32(false, A, false, B2, (short)0, c2, false, false);
            c3 = __builtin_amdgcn_wmma_f32_16x16x4_f32(false, A, false, B3, (short)0, c3, false, false);
        }
    }

    float* pout = partial + (size_t)slice * (NNODE * OUT_F);
    int rbase = i0 + ((lane >> 4) << 3);
#pragma unroll
    for (int q = 0; q < 8; ++q) {
        float* orow = pout + (rbase + q) * OUT_F + n;
        orow[0]  = c0[q];
        orow[16] = c1[q];
        orow[32] = c2[q];
        orow[48] = c3[q];
    }
}

// ---------------------------------------------------------------------------
// Kernel E: deterministic slice reduction + ELU.
// ---------------------------------------------------------------------------
__global__ void gat_reduce_elu(const float* __restrict__ partial,
                               float* __restrict__ out, int nd) {
    int idx = blockIdx.x * blockDim.x + threadIdx.x;
    float s = 0.f;
    for (int d = 0; d < nd; ++d)
        s += partial[(size_t)d * (NNODE * OUT_F) + idx];
    out[idx] = s > 0.f ? s : __expf(s) - 1.f;
}

// ---------------------------------------------------------------------------
extern "C" void kernel_launch(void* const* d_in, const int* in_sizes, int n_in,
                              void* d_out, int out_size, void* d_ws, size_t ws_size,
                              hipStream_t stream) {
    const float* input = (const float*)d_in[0];
    // d_in[1] = adj : unused by the reference -> never read.
    const float* emb   = (const float*)d_in[2];
    const float* W     = (const float*)d_in[3];
    const float* a     = (const float*)d_in[4];
    const float* iw    = (const float*)d_in[5];
    // d_in[6] = intent_b : per-column constant under softmax(axis=0) -> drops out.
    const float* beta  = (const float*)d_in[7];
    float* out = (float*)d_out;

    float* ws = (float*)d_ws;
    float* h  = ws;                              // 8192*64
    float* u  = h  + (size_t)NNODE * OUT_F;      // 8192
    float* v  = u  + NNODE;                      // 8192
    float* p  = v  + NNODE;                      // 8192
    float* r  = p  + NNODE;                      // 8192
    float* pm = r  + NNODE;                      // CSLICE*8192
    float* pz = pm + (size_t)CSLICE * NNODE;     // CSLICE*8192
    float* pd = pz + (size_t)CSLICE * NNODE;     // nd * 8192*64 partials

    // choose j-slice count from available workspace (deterministic per size)
    size_t base  = (size_t)(pd - ws);
    size_t avail = ws_size / sizeof(float);
    int nd = 1;
    if (avail >= base + 4u * NNODE * OUT_F)      nd = 4;
    else if (avail >= base + 2u * NNODE * OUT_F) nd = 2;
    int jlen = NNODE / nd;

    gat_h_kernel<<<(NNODE * OUT_F) / 256, 256, 0, stream>>>(input, W, h);
    gat_vec_kernel<<<NNODE / 256, 256, 0, stream>>>(h, emb, a, iw, beta, u, v, p);
    gat_colstats_part<<<32 * CSLICE, 256, 0, stream>>>(u, p, v, pm, pz);
    gat_colstats_merge<<<NNODE / 256, 256, 0, stream>>>(pm, pz, r);
    gat_attn_wmma_kernel<<<(NNODE / 16 / 4) * nd, 128, 0, stream>>>(h, u, p, v, r, pd, jlen, nd);
    gat_reduce_elu<<<(NNODE * OUT_F) / 256, 256, 0, stream>>>(pd, out, nd);
}